// GCN_36361193128464
// MI455X (gfx1250) — compile-verified
//
#include <hip/hip_runtime.h>
#include <hip/hip_bf16.h>
#include <math.h>

// GCN forward for MI455X (gfx1250, wave32).
//   h   = relu(spmm(A, X@W1) + b1)
//   out = log_softmax(spmm(A, h@W2) + b2)
// fp32 throughout; GEMMs use V_WMMA_F32_16X16X4_F32.

#define NNODES 100000
#define NFEAT  256
#define NHID   128
#define NCLASS 40

typedef __attribute__((ext_vector_type(2))) float v2f;
typedef __attribute__((ext_vector_type(8))) float v8f;

// ---------------------------------------------------------------------------
// Kernel 1: sorted-COO rows -> CSR row pointers.
// rp[r] = first edge index e with adj_row[e] >= r ; rp[N] = E.
// ---------------------------------------------------------------------------
__global__ void k_rowptr(const int* __restrict__ row, int* __restrict__ rp,
                         int E, int n) {
    int e = blockIdx.x * blockDim.x + threadIdx.x;
    if (e >= E) return;
    int r    = row[e];
    int prev = (e == 0) ? -1 : row[e - 1];
    for (int q = prev + 1; q <= r; ++q) rp[q] = e;
    if (e == E - 1) {
        for (int q = r + 1; q <= n; ++q) rp[q] = E;
    }
}

// ---------------------------------------------------------------------------
// Kernel 2: support = X @ W1   [N,256]x[256,128], f32 WMMA 16x16x4.
// Block: 256 thr = 8 waves; block owns 64 rows x 128 cols. Wave w owns the
// 16-wide col strip n0=16*w and computes FOUR 16x16 row-tiles with shared
// B fragments (4x W1 traffic reuse). X tile staged in LDS, stride 260
// (260 % 64 == 4 -> conflict-free strided A reads).
// ---------------------------------------------------------------------------
__global__ void __launch_bounds__(256) k_gemm1(const float* __restrict__ X,
                                               const float* __restrict__ W1,
                                               float* __restrict__ S, int n) {
    __shared__ float xs[64 * 260];

    const int m0 = blockIdx.x * 64;
    // cooperative load: 64 rows x 256 floats (64 float4 per row)
    for (int idx = threadIdx.x; idx < 64 * 64; idx += 256) {
        int r  = idx >> 6;
        int c4 = idx & 63;
        int mr = m0 + r; if (mr > n - 1) mr = n - 1;   // clamp ragged tail
        float4 v = *(const float4*)(X + (size_t)mr * NFEAT + c4 * 4);
        float* d = xs + r * 260 + c4 * 4;
        d[0] = v.x; d[1] = v.y; d[2] = v.z; d[3] = v.w;
    }
    __syncthreads();

    const int wave = threadIdx.x >> 5;
    const int lane = threadIdx.x & 31;
    const int half = lane >> 4;       // 0: K pair {0,1}; 1: K pair {2,3}
    const int l16  = lane & 15;
    const int n0   = wave * 16;

    v8f acc[4] = {};
    const float* xr[4];
#pragma unroll
    for (int t = 0; t < 4; ++t)
        xr[t] = xs + (l16 + 16 * t) * 260 + 2 * half;
    const float* wcol = W1 + (size_t)(2 * half) * NHID + n0 + l16;

    for (int k = 0; k < NFEAT; k += 4) {
        v2f b;
        b.x = wcol[(size_t)k * NHID];
        b.y = wcol[(size_t)(k + 1) * NHID];
#pragma unroll
        for (int t = 0; t < 4; ++t) {
            v2f a;
            a.x = xr[t][k];
            a.y = xr[t][k + 1];
            acc[t] = __builtin_amdgcn_wmma_f32_16x16x4_f32(
                false, a, false, b, (short)0, acc[t], false, false);
        }
    }

    // C layout: VGPR j -> row (tile_base + j + 8*half), col n0 + l16
#pragma unroll
    for (int t = 0; t < 4; ++t) {
        const int mt = m0 + 16 * t;
        float* outp = S + (size_t)(mt + 8 * half) * NHID + n0 + l16;
        if (mt + 16 <= n) {            // uniform fast path (N%16==0)
#pragma unroll
            for (int j = 0; j < 8; ++j) outp[(size_t)j * NHID] = acc[t][j];
        } else {
#pragma unroll
            for (int j = 0; j < 8; ++j)
                if (mt + 8 * half + j < n) outp[(size_t)j * NHID] = acc[t][j];
        }
    }
}

// ---------------------------------------------------------------------------
// Kernel 3: H = relu(spmm(A, S) + b1).  One wave per row, lane holds float4
// of the 128-wide row. 32-edge chunks, lane-loaded then shfl-broadcast.
// ---------------------------------------------------------------------------
__global__ void __launch_bounds__(256) k_spmm1(const int* __restrict__ rp,
                                               const int* __restrict__ col,
                                               const float* __restrict__ val,
                                               const float* __restrict__ S,
                                               const float* __restrict__ b1,
                                               float* __restrict__ H, int n) {
    const int wave = blockIdx.x * (blockDim.x >> 5) + (threadIdx.x >> 5);
    const int lane = threadIdx.x & 31;
    if (wave >= n) return;                       // uniform per wave

    const int e0 = rp[wave], e1 = rp[wave + 1];
    float4 acc = {0.f, 0.f, 0.f, 0.f};

    for (int base = e0; base < e1; base += 32) {
        int   e = base + lane;
        float v = 0.f;
        int   c = 0;
        if (e < e1) { v = val[e]; c = col[e]; }
        const int cnt = min(32, e1 - base);
        for (int j = 0; j < cnt; ++j) {
            const float vj = __shfl(v, j, 32);
            const int   cj = __shfl(c, j, 32);
            const float4 r = *(const float4*)(S + (size_t)cj * NHID + lane * 4);
            acc.x += vj * r.x; acc.y += vj * r.y;
            acc.z += vj * r.z; acc.w += vj * r.w;
        }
    }
    const float4 bb = *(const float4*)(b1 + lane * 4);
    acc.x = fmaxf(acc.x + bb.x, 0.f);
    acc.y = fmaxf(acc.y + bb.y, 0.f);
    acc.z = fmaxf(acc.z + bb.z, 0.f);
    acc.w = fmaxf(acc.w + bb.w, 0.f);
    *(float4*)(H + (size_t)wave * NHID + lane * 4) = acc;
}

// ---------------------------------------------------------------------------
// Kernel 4: T = H @ W2   [N,128]x[128,40], f32 WMMA. W2 zero-padded to 48
// cols in LDS -> 3 clean 16x16 tiles per wave. H block tile (128 rows)
// staged in LDS with stride 132 (132 % 64 == 4 -> conflict-free).
// ---------------------------------------------------------------------------
__global__ void __launch_bounds__(256) k_gemm2(const float* __restrict__ H,
                                               const float* __restrict__ W2,
                                               float* __restrict__ T, int n) {
    __shared__ float hs[128 * 132];
    __shared__ float w2s[NHID * 48];

    const int m0b = blockIdx.x * 128;
    for (int idx = threadIdx.x; idx < NHID * 48; idx += 256) {
        int k = idx / 48, c = idx - k * 48;
        w2s[idx] = (c < NCLASS) ? W2[k * NCLASS + c] : 0.f;
    }
    for (int idx = threadIdx.x; idx < 128 * 32; idx += 256) {
        int r  = idx >> 5;
        int c4 = idx & 31;
        int mr = m0b + r; if (mr > n - 1) mr = n - 1;   // clamp ragged tail
        float4 v = *(const float4*)(H + (size_t)mr * NHID + c4 * 4);
        float* d = hs + r * 132 + c4 * 4;
        d[0] = v.x; d[1] = v.y; d[2] = v.z; d[3] = v.w;
    }
    __syncthreads();

    const int wave = threadIdx.x >> 5;
    const int lane = threadIdx.x & 31;
    const int half = lane >> 4;
    const int l16  = lane & 15;
    const int m0   = m0b + wave * 16;

    const float* hrow = hs + (wave * 16 + l16) * 132 + 2 * half;

    v8f acc0 = {}, acc1 = {}, acc2 = {};
    for (int k = 0; k < NHID; k += 4) {
        v2f a;
        a.x = hrow[k];
        a.y = hrow[k + 1];
        const float* wr = w2s + (k + 2 * half) * 48 + l16;
        v2f b0, b1v, b2v;
        b0.x  = wr[0];      b0.y  = wr[48];
        b1v.x = wr[16];     b1v.y = wr[48 + 16];
        b2v.x = wr[32];     b2v.y = wr[48 + 32];
        acc0 = __builtin_amdgcn_wmma_f32_16x16x4_f32(
            false, a, false, b0, (short)0, acc0, false, false);
        acc1 = __builtin_amdgcn_wmma_f32_16x16x4_f32(
            false, a, false, b1v, (short)0, acc1, false, false);
        acc2 = __builtin_amdgcn_wmma_f32_16x16x4_f32(
            false, a, false, b2v, (short)0, acc2, false, false);
    }

    if (m0 + 16 <= n) {                 // uniform fast path
#pragma unroll
        for (int j = 0; j < 8; ++j) {
            float* tr = T + (size_t)(m0 + 8 * half + j) * NCLASS;
            tr[l16]      = acc0[j];
            tr[16 + l16] = acc1[j];
            if (l16 < 8) tr[32 + l16] = acc2[j];
        }
    } else {
#pragma unroll
        for (int j = 0; j < 8; ++j) {
            const int m = m0 + 8 * half + j;
            if (m < n) {
                float* tr = T + (size_t)m * NCLASS;
                tr[l16]      = acc0[j];
                tr[16 + l16] = acc1[j];
                if (l16 < 8) tr[32 + l16] = acc2[j];
            }
        }
    }
}

// ---------------------------------------------------------------------------
// Kernel 5: out = log_softmax(spmm(A, T) + b2). One wave per row; lanes 0-19
// each hold 2 of the 40 classes; cross-lane max / sum-exp reductions.
// ---------------------------------------------------------------------------
__global__ void __launch_bounds__(256) k_spmm2(const int* __restrict__ rp,
                                               const int* __restrict__ col,
                                               const float* __restrict__ val,
                                               const float* __restrict__ T,
                                               const float* __restrict__ b2,
                                               float* __restrict__ out, int n) {
    const int wave = blockIdx.x * (blockDim.x >> 5) + (threadIdx.x >> 5);
    const int lane = threadIdx.x & 31;
    if (wave >= n) return;

    const bool act = lane < 20;
    const int  e0 = rp[wave], e1 = rp[wave + 1];
    float a0 = 0.f, a1 = 0.f;

    for (int base = e0; base < e1; base += 32) {
        int   e = base + lane;
        float v = 0.f;
        int   c = 0;
        if (e < e1) { v = val[e]; c = col[e]; }
        const int cnt = min(32, e1 - base);
        for (int j = 0; j < cnt; ++j) {
            const float vj = __shfl(v, j, 32);
            const int   cj = __shfl(c, j, 32);
            if (act) {
                const float2 r =
                    *(const float2*)(T + (size_t)cj * NCLASS + lane * 2);
                a0 += vj * r.x;
                a1 += vj * r.y;
            }
        }
    }

    if (act) {
        a0 += b2[lane * 2];
        a1 += b2[lane * 2 + 1];
    } else {
        a0 = -INFINITY;
        a1 = -INFINITY;
    }

    // max over 40 logits
    float m = fmaxf(a0, a1);
#pragma unroll
    for (int off = 16; off >= 1; off >>= 1)
        m = fmaxf(m, __shfl_xor(m, off, 32));
    // sum of exp
    float s = act ? (__expf(a0 - m) + __expf(a1 - m)) : 0.f;
#pragma unroll
    for (int off = 16; off >= 1; off >>= 1)
        s += __shfl_xor(s, off, 32);
    const float ls = __logf(s);

    if (act) {
        out[(size_t)wave * NCLASS + lane * 2]     = a0 - m - ls;
        out[(size_t)wave * NCLASS + lane * 2 + 1] = a1 - m - ls;
    }
}

// ---------------------------------------------------------------------------
extern "C" void kernel_launch(void* const* d_in, const int* in_sizes, int n_in,
                              void* d_out, int out_size, void* d_ws,
                              size_t ws_size, hipStream_t stream) {
    const float* x       = (const float*)d_in[0];
    const int*   adj_row = (const int*)d_in[1];
    const int*   adj_col = (const int*)d_in[2];
    const float* adj_val = (const float*)d_in[3];
    // d_in[4] = i (unused scalar)
    const float* W1 = (const float*)d_in[5];
    const float* b1 = (const float*)d_in[6];
    const float* W2 = (const float*)d_in[7];
    const float* b2 = (const float*)d_in[8];
    float* out = (float*)d_out;

    const int n = in_sizes[0] / NFEAT;  // 100000
    const int E = in_sizes[1];          // 3200000

    // workspace layout: support[N*128] | H[N*128] | T[N*40] | rp[N+1]
    float* support = (float*)d_ws;
    float* H       = support + (size_t)n * NHID;
    float* T       = H + (size_t)n * NHID;
    int*   rp      = (int*)(T + (size_t)n * NCLASS);

    k_rowptr<<<(E + 255) / 256, 256, 0, stream>>>(adj_row, rp, E, n);
    k_gemm1<<<(n + 63) / 64, 256, 0, stream>>>(x, W1, support, n);
    k_spmm1<<<(n + 7) / 8, 256, 0, stream>>>(rp, adj_col, adj_val, support, b1,
                                             H, n);
    k_gemm2<<<(n + 127) / 128, 256, 0, stream>>>(H, W2, T, n);
    k_spmm2<<<(n + 7) / 8, 256, 0, stream>>>(rp, adj_col, adj_val, T, b2, out,
                                             n);
}